// PointPillarsPreProcess_82884278879144
// MI455X (gfx1250) — compile-verified
//
#include <hip/hip_runtime.h>
#include <hip/hip_bf16.h>
#include <cstdint>

// ---------------- problem constants (from reference) ----------------
constexpr int B        = 4;
constexpr int N        = 300000;
constexpr int C        = 5;
constexpr int BN       = B * N;            // 1,200,000 points
constexpr int GX       = 432;
constexpr int GY       = 496;
constexpr int GZ       = 1;
constexpr int GXY      = GX * GY;          // 214,272
constexpr int NVOX     = GX * GY * GZ;     // 214,272
constexpr int MAXV     = 40000;
constexpr int MAXP     = 30;
constexpr int SCAN_ELE = 1024;             // flags per scan block (256 thr x 4)
constexpr int NBLK     = (NVOX + SCAN_ELE - 1) / SCAN_ELE;   // 210 (<=256)
constexpr int BSUM_STR = 256;

constexpr size_t FEATS_ELEMS  = (size_t)B * MAXV * MAXP * C; // 24,000,000 f32
constexpr size_t COORDS_ELEMS = (size_t)B * MAXV * 4;        // 640,000 i32

// total 16B chunks in the whole points tensor: BN*C*4 / 16 = 1,500,000 (exact)
constexpr long TOT_CHUNKS = (long)BN * C * 4 / 16;

// pc_range / voxel constants (+ compile-time reciprocals; 1/0.16=6.25, 1/4=0.25 exact)
constexpr float PX_LO = 0.0f,   PY_LO = -39.68f, PZ_LO = -3.0f;
constexpr float INV_VSX = 1.0f / 0.16f;   // 6.25 exact
constexpr float INV_VSY = 1.0f / 0.16f;
constexpr float INV_VSZ = 1.0f / 4.0f;    // 0.25 exact
constexpr float INV_SPX = 1.0f / 69.12f;
constexpr float INV_SPY = 1.0f / 79.36f;
constexpr float INV_SPZ = 1.0f / 4.0f;

// ---------------- pass 1: mark occupied voxels (12B/point via async b32) ----------------
__global__ __launch_bounds__(256) void k_mark(const float* __restrict__ pts,
                                              int* __restrict__ occ) {
    __shared__ __align__(16) float sm[256 * 3];
    const int t   = threadIdx.x;
    const int gid = blockIdx.x * 256 + t;
    const int cg  = gid < BN ? gid : BN - 1;          // clamp keeps all lanes in-bounds
    const float* gp = pts + (size_t)cg * C;
    const unsigned lp = (unsigned)(uintptr_t)(&sm[t * 3]);

    asm volatile(
        "global_load_async_to_lds_b32 %0, %1, off\n\t"
        "global_load_async_to_lds_b32 %0, %1, off offset:4\n\t"
        "global_load_async_to_lds_b32 %0, %1, off offset:8"
        :: "v"(lp), "v"(gp) : "memory");
    asm volatile("s_wait_asynccnt 0" ::: "memory");

    const float px = sm[t * 3 + 0];
    const float py = sm[t * 3 + 1];
    const float pz = sm[t * 3 + 2];
    if (gid >= BN) return;

    const int ix = (int)floorf((px - PX_LO) * INV_VSX);
    const int iy = (int)floorf((py - PY_LO) * INV_VSY);
    const int iz = (int)floorf((pz - PZ_LO) * INV_VSZ);
    if (ix < 0 || ix >= GX || iy < 0 || iy >= GY || iz < 0 || iz >= GZ) return;

    const int b   = cg / N;
    const int vid = (iz * GY + iy) * GX + ix;
    occ[(size_t)b * NVOX + vid] = 1;                  // benign race: all write 1
}

// ---------------- pass 2a: per-block sums of occupancy flags ----------------
__global__ __launch_bounds__(256) void k_scan1(const int* __restrict__ occ,
                                               int* __restrict__ bsum) {
    const int b = blockIdx.y, blk = blockIdx.x, t = threadIdx.x;
    const int base = blk * SCAN_ELE + t * 4;
    int s = 0;
#pragma unroll
    for (int k = 0; k < 4; ++k) {
        const int v = base + k;
        if (v < NVOX) s += occ[(size_t)b * NVOX + v];
    }
    __shared__ int sh[256];
    sh[t] = s;
    __syncthreads();
    for (int off = 128; off > 0; off >>= 1) {
        if (t < off) sh[t] += sh[t + off];
        __syncthreads();
    }
    if (t == 0) bsum[b * BSUM_STR + blk] = sh[0];
}

// ---------------- pass 2b: parallel exclusive scan of block sums ----------------
__global__ __launch_bounds__(256) void k_scan2(int* __restrict__ bsum) {
    const int b = blockIdx.x, t = threadIdx.x;
    const int v = (t < NBLK) ? bsum[b * BSUM_STR + t] : 0;
    __shared__ int sh[256];
    sh[t] = v;
    __syncthreads();
    for (int off = 1; off < 256; off <<= 1) {     // Hillis-Steele inclusive
        const int x = (t >= off) ? sh[t - off] : 0;
        __syncthreads();
        sh[t] += x;
        __syncthreads();
    }
    if (t < NBLK) bsum[b * BSUM_STR + t] = sh[t] - v;   // exclusive
}

// ---------------- pass 2c: slot assignment + coords ----------------
__global__ __launch_bounds__(256) void k_scan3(const int* __restrict__ occ,
                                               const int* __restrict__ bsum,
                                               int* __restrict__ slotmap,
                                               int* __restrict__ coords) {
    const int b = blockIdx.y, blk = blockIdx.x, t = threadIdx.x;
    const int base = blk * SCAN_ELE + t * 4;
    int f[4];
    int s = 0;
#pragma unroll
    for (int k = 0; k < 4; ++k) {
        const int v = base + k;
        f[k] = (v < NVOX) ? occ[(size_t)b * NVOX + v] : 0;
        s += f[k];
    }
    __shared__ int sh[256];
    sh[t] = s;
    __syncthreads();
    for (int off = 1; off < 256; off <<= 1) {
        const int x = (t >= off) ? sh[t - off] : 0;
        __syncthreads();
        sh[t] += x;
        __syncthreads();
    }
    int rank = bsum[b * BSUM_STR + blk] + sh[t] - s;   // exclusive prefix for this thread
#pragma unroll
    for (int k = 0; k < 4; ++k) {
        const int v = base + k;
        if (v < NVOX) {
            const bool take = (f[k] != 0) && (rank < MAXV);
            slotmap[(size_t)b * NVOX + v] = take ? rank : -1;
            if (take) {
                const int vz = v / GXY;
                const int rem = v % GXY;
                int4* cp = (int4*)(coords + ((size_t)(b * MAXV + rank)) * 4); // 16B aligned
                *cp = make_int4(b, vz, rem / GX, rem % GX);
            }
            rank += f[k];
        }
    }
}

// ---------------- pass 3: scatter normalized features ----------------
// Stage 256 points (5120B) per block as 320 aligned 16B chunks via async b128.
__global__ __launch_bounds__(256) void k_scatter(const float* __restrict__ pts,
                                                 const int* __restrict__ slotmap,
                                                 int* __restrict__ cnt,
                                                 float* __restrict__ feats) {
    __shared__ __align__(16) float sm[256 * 5];       // 5120 bytes = 320 chunks
    const int t   = threadIdx.x;
    const int gid = blockIdx.x * 256 + t;
    const unsigned smBase = (unsigned)(uintptr_t)sm;
    const long blockChunk0 = (long)blockIdx.x * 320;

#pragma unroll
    for (int i = t; i < 320; i += 256) {              // 2nd iter: only t<64 active
        long g = blockChunk0 + i;
        if (g > TOT_CHUNKS - 1) g = TOT_CHUNKS - 1;   // chunk-granular clamp (in-bounds)
        const float* gp = pts + g * 4;                // 16B per chunk
        const unsigned lp = smBase + (unsigned)i * 16u;
        asm volatile("global_load_async_to_lds_b128 %0, %1, off"
                     :: "v"(lp), "v"(gp) : "memory");
    }
    asm volatile("s_wait_asynccnt 0" ::: "memory");
    __syncthreads();                                  // publish cross-lane chunks

    const float px = sm[t * 5 + 0];
    const float py = sm[t * 5 + 1];
    const float pz = sm[t * 5 + 2];
    const float p3 = sm[t * 5 + 3];
    const float p4 = sm[t * 5 + 4];
    if (gid >= BN) return;

    const int ix = (int)floorf((px - PX_LO) * INV_VSX);
    const int iy = (int)floorf((py - PY_LO) * INV_VSY);
    const int iz = (int)floorf((pz - PZ_LO) * INV_VSZ);
    if (ix < 0 || ix >= GX || iy < 0 || iy >= GY || iz < 0 || iz >= GZ) return;

    const int b   = gid / N;
    const int vid = (iz * GY + iy) * GX + ix;
    const int sl  = slotmap[(size_t)b * NVOX + vid];  // L2-resident table
    if (sl < 0) return;

    const int r = atomicAdd(&cnt[b * MAXV + sl], 1);
    if (r >= MAXP) return;

    const size_t o = ((size_t)(b * MAXV + sl) * MAXP + r) * (size_t)C;
    feats[o + 0] = (px - PX_LO) * INV_SPX;
    feats[o + 1] = (py - PY_LO) * INV_SPY;
    feats[o + 2] = (pz - PZ_LO) * INV_SPZ;
    feats[o + 3] = p3;
    feats[o + 4] = p4;
}

// ---------------- launcher ----------------
extern "C" void kernel_launch(void* const* d_in, const int* in_sizes, int n_in,
                              void* d_out, int out_size, void* d_ws, size_t ws_size,
                              hipStream_t stream) {
    const float* pts = (const float*)d_in[0];

    float* feats  = (float*)d_out;
    int*   coords = (int*)((float*)d_out + FEATS_ELEMS);

    // workspace layout
    int* occ     = (int*)d_ws;                       // B*NVOX
    int* slotmap = occ + (size_t)B * NVOX;           // B*NVOX
    int* bsum    = slotmap + (size_t)B * NVOX;       // B*BSUM_STR
    int* cnt     = bsum + (size_t)B * BSUM_STR;      // B*MAXV

    // deterministic re-init every call (graph-capture safe)
    hipMemsetAsync(occ,    0,    (size_t)B * NVOX * sizeof(int), stream);
    hipMemsetAsync(cnt,    0,    (size_t)B * MAXV * sizeof(int), stream);
    hipMemsetAsync(feats,  0,    FEATS_ELEMS * sizeof(float),    stream);
    hipMemsetAsync(coords, 0xFF, COORDS_ELEMS * sizeof(int),     stream);  // -1

    const int ptBlocks = (BN + 255) / 256;
    k_mark<<<ptBlocks, 256, 0, stream>>>(pts, occ);

    dim3 sg(NBLK, B);
    k_scan1<<<sg, 256, 0, stream>>>(occ, bsum);
    k_scan2<<<B, 256, 0, stream>>>(bsum);
    k_scan3<<<sg, 256, 0, stream>>>(occ, bsum, slotmap, coords);

    k_scatter<<<ptBlocks, 256, 0, stream>>>(pts, slotmap, cnt, feats);
}